// Model_39676907881128
// MI455X (gfx1250) — compile-verified
//
#include <hip/hip_runtime.h>

// CDNA5 / gfx1250, wave32. f32 WMMA 16x16x4 fragments:
//   A (16x4 f32): lane L -> M = L%16, K pair = 2*(L/16)   (a.x=K, a.y=K+1)
//   B (4x16 f32): lane L -> N = L%16, K pair = 2*(L/16)   (b.x=K, b.y=K+1)
//   C/D (16x16):  lane L, vgpr v -> M = v + 8*(L/16), N = L%16
typedef __attribute__((ext_vector_type(2))) float v2f;
typedef __attribute__((ext_vector_type(8))) float v8f;

#define WMMA_F32(a, b, c) \
    __builtin_amdgcn_wmma_f32_16x16x4_f32(false, (a), false, (b), (short)0, (c), false, false)

// ---------------------------------------------------------------------------
// Y[M x 64] = X[M x 64] @ W^T + bias.   One wave = 16 rows x all 64 cols.
// grid.x * 4 waves must equal M/16.
// ---------------------------------------------------------------------------
__global__ __launch_bounds__(128) void linear64_kernel(const float* __restrict__ X,
                                                       const float* __restrict__ W,
                                                       const float* __restrict__ bias,
                                                       float* __restrict__ Y)
{
    const int lane = threadIdx.x & 31;
    const int wave = threadIdx.x >> 5;
    const int half = lane >> 4;
    const int l16  = lane & 15;
    const int tile = blockIdx.x * 4 + wave;

    const float* xrow = X + (size_t)tile * 16 * 64 + l16 * 64 + 2 * half;

    v8f acc0 = {}, acc1 = {}, acc2 = {}, acc3 = {};
#pragma unroll
    for (int k0 = 0; k0 < 64; k0 += 4) {
        v2f a = *(const v2f*)(xrow + k0);
        // B[e,o] = W[o,e]  ->  b.x = W[n, k], b.y = W[n, k+1] (contiguous pair)
        const float* wb = W + (size_t)l16 * 64 + k0 + 2 * half;
        v2f b0 = *(const v2f*)(wb);
        v2f b1 = *(const v2f*)(wb + 16 * 64);
        v2f b2 = *(const v2f*)(wb + 32 * 64);
        v2f b3 = *(const v2f*)(wb + 48 * 64);
        acc0 = WMMA_F32(a, b0, acc0);
        acc1 = WMMA_F32(a, b1, acc1);
        acc2 = WMMA_F32(a, b2, acc2);
        acc3 = WMMA_F32(a, b3, acc3);
    }

    const float bb0 = bias[l16];
    const float bb1 = bias[16 + l16];
    const float bb2 = bias[32 + l16];
    const float bb3 = bias[48 + l16];
#pragma unroll
    for (int v = 0; v < 8; ++v) {
        const int row = tile * 16 + v + 8 * half;
        float* yr = Y + (size_t)row * 64 + l16;
        yr[0]  = acc0[v] + bb0;
        yr[16] = acc1[v] + bb1;
        yr[32] = acc2[v] + bb2;
        yr[48] = acc3[v] + bb3;
    }
}

// ---------------------------------------------------------------------------
// qk[j,b,c] = (1/8) * sum_e q[j,b,e] * k[j,c,e].   16 waves total (grid 4 x 128).
// ---------------------------------------------------------------------------
__global__ __launch_bounds__(128) void qk_kernel(const float* __restrict__ qp, // [4,64,64]
                                                 const float* __restrict__ kp, // [4,64,64]
                                                 float* __restrict__ qk)       // [4,64,64]
{
    const int lane  = threadIdx.x & 31;
    const int wave  = threadIdx.x >> 5;
    const int half  = lane >> 4;
    const int l16   = lane & 15;
    const int wg    = blockIdx.x * 4 + wave;   // 0..15
    const int j     = wg >> 2;
    const int mtile = wg & 3;

    const float* arow = qp + (size_t)j * 4096 + mtile * 16 * 64 + l16 * 64 + 2 * half;
    const float* Kj   = kp + (size_t)j * 4096;

    v8f acc0 = {}, acc1 = {}, acc2 = {}, acc3 = {};
#pragma unroll
    for (int k0 = 0; k0 < 64; k0 += 4) {
        v2f a = *(const v2f*)(arow + k0);
        // B[e,c] = k[j,c,e] -> load row c of k_j, contiguous K pair
        const float* kb = Kj + (size_t)l16 * 64 + k0 + 2 * half;
        v2f b0 = *(const v2f*)(kb);
        v2f b1 = *(const v2f*)(kb + 16 * 64);
        v2f b2 = *(const v2f*)(kb + 32 * 64);
        v2f b3 = *(const v2f*)(kb + 48 * 64);
        acc0 = WMMA_F32(a, b0, acc0);
        acc1 = WMMA_F32(a, b1, acc1);
        acc2 = WMMA_F32(a, b2, acc2);
        acc3 = WMMA_F32(a, b3, acc3);
    }

    const float s = 0.125f; // 1/sqrt(64)
#pragma unroll
    for (int v = 0; v < 8; ++v) {
        const int row = mtile * 16 + v + 8 * half;
        float* o = qk + (size_t)j * 4096 + row * 64 + l16;
        o[0]  = acc0[v] * s;
        o[16] = acc1[v] * s;
        o[32] = acc2[v] * s;
        o[48] = acc3[v] * s;
    }
}

// ---------------------------------------------------------------------------
// attn[i,j,a,b,c] = softmax_b( qk[j,b,c] + mask[i,a,b] ), stored flat so the
// buffer IS the row-major [64,1024,64] A-matrix of the final GEMM.
// grid = 1024 blocks (one per (i,j,a)), 64 threads (one per c).
// ---------------------------------------------------------------------------
__global__ __launch_bounds__(64) void softmax_kernel(const float* __restrict__ qk,   // [4,64,64] (j,b,c)
                                                     const float* __restrict__ mask, // [4,64,64] (i,a,b)
                                                     float* __restrict__ attn)       // [4,4,64,64,64]
{
    const int c   = threadIdx.x;      // 0..63
    const int bid = blockIdx.x;       // (i*4+j)*64 + a
    const int a   = bid & 63;
    const int j   = (bid >> 6) & 3;
    const int i   = bid >> 8;

    const float* qkj = qk + (size_t)j * 4096 + c;        // qk[j,b,c], stride 64 over b
    const float* mia = mask + (size_t)i * 4096 + a * 64; // mask[i,a,b]

    float denom = 0.f;
#pragma unroll 8
    for (int b = 0; b < 64; ++b)
        denom += __expf(qkj[b * 64] + mia[b]);
    const float inv = 1.0f / denom;

    float* o = attn + (size_t)bid * 4096 + c;            // bid*4096 + b*64 + c
#pragma unroll 8
    for (int b = 0; b < 64; ++b)
        o[b * 64] = __expf(qkj[b * 64] + mia[b]) * inv;
}

// ---------------------------------------------------------------------------
// out[h,p,c'] = sum_d attn[h,p,d] * v3[h,d,c'],  h<64, p<1024, c'<1024, d<64.
// One wave = 16 (M) x 64 (N) output strip, A fragment reused over 4 B subtiles.
// grid = 64 h * 64 mtiles * 4 nblks = 16384 blocks x 128 threads (4 waves).
// ---------------------------------------------------------------------------
__global__ __launch_bounds__(128) void attn_out_kernel(const float* __restrict__ attn, // [64,1024,64]
                                                       const float* __restrict__ v3,   // [64,64,1024]
                                                       float* __restrict__ out)        // [64,1024,1024]
{
    const int lane = threadIdx.x & 31;
    const int wave = threadIdx.x >> 5;
    const int half = lane >> 4;
    const int l16  = lane & 15;

    const int bid   = blockIdx.x;
    const int h     = bid >> 8;
    const int rem   = bid & 255;
    const int mtile = rem >> 2;
    const int nblk  = rem & 3;
    const int ncol0 = nblk * 256 + wave * 64;

    const float* A = attn + (size_t)h * 65536 + (size_t)mtile * 16 * 64; // [16 x 64], ld 64
    const float* B = v3   + (size_t)h * 65536 + ncol0;                   // [64 x 1024], ld 1024
    float*       C = out  + (size_t)h * 1024 * 1024 + (size_t)mtile * 16 * 1024 + ncol0;

    const float* arow = A + l16 * 64 + 2 * half;

    v8f acc0 = {}, acc1 = {}, acc2 = {}, acc3 = {};
#pragma unroll
    for (int k0 = 0; k0 < 64; k0 += 4) {
        v2f a = *(const v2f*)(arow + k0);
        const float* brow = B + (size_t)(k0 + 2 * half) * 1024 + l16;    // row K, col l16 (coalesced)
        v2f b0, b1, b2, b3;
        b0.x = brow[0];  b0.y = brow[1024];
        b1.x = brow[16]; b1.y = brow[16 + 1024];
        b2.x = brow[32]; b2.y = brow[32 + 1024];
        b3.x = brow[48]; b3.y = brow[48 + 1024];
        acc0 = WMMA_F32(a, b0, acc0);
        acc1 = WMMA_F32(a, b1, acc1);
        acc2 = WMMA_F32(a, b2, acc2);
        acc3 = WMMA_F32(a, b3, acc3);
    }

#pragma unroll
    for (int v = 0; v < 8; ++v) {
        const int row = v + 8 * half;
        float* crow = C + (size_t)row * 1024 + l16;
        crow[0]  = acc0[v];
        crow[16] = acc1[v];
        crow[32] = acc2[v];
        crow[48] = acc3[v];
    }
}

// ---------------------------------------------------------------------------
// inputs: query, key, value, mask, Wq, bq, Wk, bk, Wv, bv (all f32)
// workspace layout (needs ~33.8 MB):
//   [0,64K)        qp  [256x64]
//   [64K,128K)     kp  [256x64]
//   [128K,192K)    qk  [4x64x64]
//   [192K,192K+16M)      vp   [65536x64]
//   [192K+16M,192K+32M)  attn [64x1024x64]
// ---------------------------------------------------------------------------
extern "C" void kernel_launch(void* const* d_in, const int* in_sizes, int n_in,
                              void* d_out, int out_size, void* d_ws, size_t ws_size,
                              hipStream_t stream)
{
    (void)in_sizes; (void)n_in; (void)out_size; (void)ws_size;

    const float* query = (const float*)d_in[0];
    const float* key   = (const float*)d_in[1];
    const float* value = (const float*)d_in[2];
    const float* mask  = (const float*)d_in[3];
    const float* Wq    = (const float*)d_in[4];
    const float* bq    = (const float*)d_in[5];
    const float* Wk    = (const float*)d_in[6];
    const float* bk    = (const float*)d_in[7];
    const float* Wv    = (const float*)d_in[8];
    const float* bv    = (const float*)d_in[9];
    float* out = (float*)d_out;

    char*  ws   = (char*)d_ws;
    float* qp   = (float*)(ws);
    float* kp   = (float*)(ws + (64 << 10));
    float* qk   = (float*)(ws + (128 << 10));
    float* vp   = (float*)(ws + (192 << 10));
    float* attn = (float*)(ws + (192 << 10) + (16 << 20));

    // q, k projections: 256 rows -> 16 tiles -> 4 blocks of 4 waves
    linear64_kernel<<<4, 128, 0, stream>>>(query, Wq, bq, qp);
    linear64_kernel<<<4, 128, 0, stream>>>(key,   Wk, bk, kp);
    // v projection: 65536 rows -> 4096 tiles -> 1024 blocks
    linear64_kernel<<<1024, 128, 0, stream>>>(value, Wv, bv, vp);
    // scores
    qk_kernel<<<4, 128, 0, stream>>>(qp, kp, qk);
    // softmax over b, materialized in final-GEMM A layout
    softmax_kernel<<<1024, 64, 0, stream>>>(qk, mask, attn);
    // 64-batch [1024x64]@[64x1024] -> 256 MB output (HBM-write bound)
    attn_out_kernel<<<16384, 128, 0, stream>>>(attn, vp, out);
}